// NEBLoss_32581621907990
// MI455X (gfx1250) — compile-verified
//
#include <hip/hip_runtime.h>
#include <hip/hip_bf16.h>
#include <math.h>

// ---------------------------------------------------------------------------
// NEB loss for MI455X (gfx1250, wave32).
//   loss_i        = -logsoftmax(output)_[i,t_i] * w[t_i]
//   P             = PCA2(embed) -> norm_embed (8192 x 2)
//   sim           = P P^T  (rank 2 -> never materialized)
//   top-5/row     -> e_i = sum exp(sim), w_i = sum exp(sim)*loss_j
//   result        = sum(ex_i*el_i) / (GAMMA * sum ex_i),  el_i = w_i*N/S
// Gram matrix (512x512, K=8192) computed with V_WMMA_F32_16X16X4_F32.
// ---------------------------------------------------------------------------

typedef __attribute__((ext_vector_type(2))) float v2f;
typedef __attribute__((ext_vector_type(8))) float v8f;

#define GAMMA_C 0.9f

// ---------------- K0: zero the argmax keys ---------------------------------
__global__ void init_kernel(unsigned long long* keys) {
  if (threadIdx.x < 2) keys[threadIdx.x] = 0ull;
}

// ---------------- K1: per-row weighted NLL loss ----------------------------
__global__ void __launch_bounds__(256)
loss_kernel(const float* __restrict__ logits, const int* __restrict__ tgt,
            const float* __restrict__ cw, float* __restrict__ loss,
            int N, int C) {
  __shared__ float red[256];
  const int row = blockIdx.x;
  const int t = threadIdx.x;
  const float* x = logits + (size_t)row * C;

  float m = -INFINITY;
  for (int c = t; c < C; c += 256) m = fmaxf(m, x[c]);
  red[t] = m; __syncthreads();
  for (int s = 128; s > 0; s >>= 1) {
    if (t < s) red[t] = fmaxf(red[t], red[t + s]);
    __syncthreads();
  }
  m = red[0]; __syncthreads();

  float se = 0.f;
  for (int c = t; c < C; c += 256) se += expf(x[c] - m);
  red[t] = se; __syncthreads();
  for (int s = 128; s > 0; s >>= 1) {
    if (t < s) red[t] += red[t + s];
    __syncthreads();
  }
  if (t == 0) {
    int tg = tgt[row];
    float lp = x[tg] - m - logf(red[0]);
    loss[row] = -lp * cw[tg];
  }
}

// ---------------- K2: column means of embed --------------------------------
__global__ void colmean_kernel(const float* __restrict__ X, float* __restrict__ mu,
                               int N, int D) {
  int c = blockIdx.x * blockDim.x + threadIdx.x;
  if (c >= D) return;
  float s = 0.f;
  for (int i = 0; i < N; ++i) s += X[(size_t)i * D + c];
  mu[c] = s / (float)N;
}

// ---------------- K3: Gram = X^T X - N mu mu^T  via f32 WMMA ---------------
// One wave per 16x16 tile of G.  A[m,k] = X[k, tm+m], B[k,n] = X[k, tn+n].
// 32-bit A 16x4 layout: lane half h = lane>>4, component v -> K = v + 2h.
// B 4x16: component v, half h -> row K = v + 2h, col N = lane&15.
// C/D: component r, half h -> M = r + 8h, N = lane&15.
__global__ void __launch_bounds__(256)
gram_kernel(const float* __restrict__ X, const float* __restrict__ mu,
            float* __restrict__ G, int N, int D) {
  const int lane = threadIdx.x & 31;
  const int wave = threadIdx.x >> 5;
  const int tile = blockIdx.x * 8 + wave;
  const int tm = (tile >> 5) << 4;   // 32 tiles per row of G (D=512)
  const int tn = (tile & 31) << 4;
  const int half = lane >> 4;
  const int mn = lane & 15;
  const int colA = tm + mn;
  const int colB = tn + mn;

  v8f acc = {0.f, 0.f, 0.f, 0.f, 0.f, 0.f, 0.f, 0.f};

  for (int k = 0; k < N; k += 4) {
    const int r0 = k + 2 * half;
    const float* xr0 = X + (size_t)r0 * D;
    const float* xr1 = xr0 + D;
    if (k + 8 < N) {  // pull next K-slab toward the caches (global_prefetch_b8)
      __builtin_prefetch(xr0 + (size_t)4 * D + colA, 0, 1);
      __builtin_prefetch(xr0 + (size_t)4 * D + colB, 0, 1);
    }
    v2f a, b;
    a[0] = xr0[colA]; a[1] = xr1[colA];
    b[0] = xr0[colB]; b[1] = xr1[colB];
    acc = __builtin_amdgcn_wmma_f32_16x16x4_f32(
        /*neg_a=*/false, a, /*neg_b=*/false, b,
        /*c_mod=*/(short)0, acc, /*reuse_a=*/false, /*reuse_b=*/false);
  }

  const float mun = mu[colB];
  #pragma unroll
  for (int r = 0; r < 8; ++r) {
    const int M = tm + r + 8 * half;
    G[(size_t)M * D + colB] = acc[r] - (float)N * mu[M] * mun;
  }
}

// ---------------- K4: top-2 eigenvectors of G (power iter + deflation) -----
__global__ void __launch_bounds__(512)
pca_kernel(const float* __restrict__ G, const float* __restrict__ mu,
           float* __restrict__ v1, float* __restrict__ v2,
           float* __restrict__ muv, int D) {
  __shared__ float sv[512];
  __shared__ float s1[512];
  __shared__ float red[512];
  const int t = threadIdx.x;
  const float* g = G + (size_t)t * D;

  // --- first eigenvector ---
  sv[t] = 1.0f + 0.5f * sinf((float)(t + 1));
  __syncthreads();
  for (int it = 0; it < 256; ++it) {
    float acc = 0.f;
    for (int k = 0; k < D; ++k) acc = fmaf(g[k], sv[k], acc);
    red[t] = acc * acc; __syncthreads();
    for (int s = 256; s > 0; s >>= 1) { if (t < s) red[t] += red[t + s]; __syncthreads(); }
    float nrm = sqrtf(red[0]); __syncthreads();
    sv[t] = (nrm > 0.f) ? acc / nrm : 0.f;
    __syncthreads();
  }
  s1[t] = sv[t];
  __syncthreads();

  // --- second eigenvector (deflate against s1 every iteration) ---
  sv[t] = 1.0f + 0.5f * cosf((float)(3 * t + 7));
  __syncthreads();
  for (int it = 0; it < 256; ++it) {
    float acc = 0.f;
    for (int k = 0; k < D; ++k) acc = fmaf(g[k], sv[k], acc);
    red[t] = s1[t] * acc; __syncthreads();
    for (int s = 256; s > 0; s >>= 1) { if (t < s) red[t] += red[t + s]; __syncthreads(); }
    float d = red[0]; __syncthreads();
    acc -= d * s1[t];
    red[t] = acc * acc; __syncthreads();
    for (int s = 256; s > 0; s >>= 1) { if (t < s) red[t] += red[t + s]; __syncthreads(); }
    float nrm = sqrtf(red[0]); __syncthreads();
    sv[t] = (nrm > 0.f) ? acc / nrm : 0.f;
    __syncthreads();
  }
  v1[t] = s1[t];
  v2[t] = sv[t];

  red[t] = mu[t] * s1[t]; __syncthreads();
  for (int s = 256; s > 0; s >>= 1) { if (t < s) red[t] += red[t + s]; __syncthreads(); }
  if (t == 0) muv[0] = red[0];
  __syncthreads();
  red[t] = mu[t] * sv[t]; __syncthreads();
  for (int s = 256; s > 0; s >>= 1) { if (t < s) red[t] += red[t + s]; __syncthreads(); }
  if (t == 0) muv[1] = red[0];
}

// ---------------- K5: projections + per-column |argmax| keys ---------------
__global__ void __launch_bounds__(128)
project_kernel(const float* __restrict__ X, const float* __restrict__ v1,
               const float* __restrict__ v2, const float* __restrict__ muv,
               float* __restrict__ proj, unsigned long long* __restrict__ keys,
               int N, int D) {
  __shared__ float sv1[512], sv2[512];
  __shared__ float r1[128], r2[128];
  const int t = threadIdx.x;
  for (int k = t; k < D; k += 128) { sv1[k] = v1[k]; sv2[k] = v2[k]; }
  __syncthreads();

  const int row = blockIdx.x;
  const float* x = X + (size_t)row * D;
  float d1 = 0.f, d2 = 0.f;
  for (int k = t; k < D; k += 128) {
    float xv = x[k];
    d1 = fmaf(xv, sv1[k], d1);
    d2 = fmaf(xv, sv2[k], d2);
  }
  r1[t] = d1; r2[t] = d2; __syncthreads();
  for (int s = 64; s > 0; s >>= 1) {
    if (t < s) { r1[t] += r1[t + s]; r2[t] += r2[t + s]; }
    __syncthreads();
  }
  if (t == 0) {
    float p0 = r1[0] - muv[0];
    float p1 = r2[0] - muv[1];
    proj[(size_t)row * 2 + 0] = p0;
    proj[(size_t)row * 2 + 1] = p1;
    // key = |p| bits (monotone for non-negative floats) << 32 | (N-1-row)
    // so ties resolve to the SMALLEST row index, matching jnp.argmax.
    unsigned long long k0 =
        ((unsigned long long)__float_as_uint(fabsf(p0)) << 32) |
        (unsigned)(N - 1 - row);
    unsigned long long k1 =
        ((unsigned long long)__float_as_uint(fabsf(p1)) << 32) |
        (unsigned)(N - 1 - row);
    atomicMax(&keys[0], k0);
    atomicMax(&keys[1], k1);
  }
}

// ---------------- K6: norm_embed = (1 + sign*proj/||row||)/2 ---------------
__global__ void normembed_kernel(const float* __restrict__ proj,
                                 const unsigned long long* __restrict__ keys,
                                 float* __restrict__ nrm, int N) {
  int i = blockIdx.x * blockDim.x + threadIdx.x;
  if (i >= N) return;
  int i0 = N - 1 - (int)(keys[0] & 0xffffffffu);
  int i1 = N - 1 - (int)(keys[1] & 0xffffffffu);
  float s0 = (proj[(size_t)i0 * 2 + 0] >= 0.f) ? 1.f : -1.f;
  float s1 = (proj[(size_t)i1 * 2 + 1] >= 0.f) ? 1.f : -1.f;
  float p0 = proj[(size_t)i * 2 + 0] * s0;
  float p1 = proj[(size_t)i * 2 + 1] * s1;
  float r = sqrtf(p0 * p0 + p1 * p1);
  float inv = (r > 0.f) ? 1.f / r : 0.f;
  nrm[(size_t)i * 2 + 0] = 0.5f * (1.f + p0 * inv);
  nrm[(size_t)i * 2 + 1] = 0.5f * (1.f + p1 * inv);
}

// ---------------- K7: fused rank-2 sim + top-5 + exp sums ------------------
// P (8192x2) and loss (8192) live in LDS (96 KB of the 320 KB WGP LDS).
__global__ void __launch_bounds__(256)
topk_kernel(const float* __restrict__ nrm, const float* __restrict__ loss,
            float* __restrict__ e, float* __restrict__ w, int N) {
  extern __shared__ float smem[];
  float* sx = smem;
  float* sy = smem + N;
  float* sl = smem + 2 * N;
  for (int j = threadIdx.x; j < N; j += 256) {
    sx[j] = nrm[(size_t)j * 2 + 0];
    sy[j] = nrm[(size_t)j * 2 + 1];
    sl[j] = loss[j];
  }
  __syncthreads();

  const int i = blockIdx.x * 256 + threadIdx.x;
  const float xi = sx[i], yi = sy[i];

  float v0 = -INFINITY, v1 = -INFINITY, v2 = -INFINITY, v3 = -INFINITY, v4 = -INFINITY;
  int j0 = 0, j1 = 0, j2 = 0, j3 = 0, j4 = 0;

  for (int j = 0; j < N; ++j) {
    float s = fmaf(xi, sx[j], yi * sy[j]);
    if (s > v4) {                       // strict: ties keep the earlier index
      v4 = s; j4 = j;
      if (v4 > v3) { float tv = v3; v3 = v4; v4 = tv; int tj = j3; j3 = j4; j4 = tj; }
      if (v3 > v2) { float tv = v2; v2 = v3; v3 = tv; int tj = j2; j2 = j3; j3 = tj; }
      if (v2 > v1) { float tv = v1; v1 = v2; v2 = tv; int tj = j1; j1 = j2; j2 = tj; }
      if (v1 > v0) { float tv = v0; v0 = v1; v1 = tv; int tj = j0; j0 = j1; j1 = tj; }
    }
  }
  // NEB_TAU == 1.0
  float e0 = expf(v0), e1 = expf(v1), e2 = expf(v2), e3 = expf(v3), e4 = expf(v4);
  e[i] = e0 + e1 + e2 + e3 + e4;
  w[i] = e0 * sl[j0] + e1 * sl[j1] + e2 * sl[j2] + e3 * sl[j3] + e4 * sl[j4];
}

// ---------------- K8: deterministic reductions (S, sum loss) ---------------
__global__ void __launch_bounds__(1024)
reduce_kernel(const float* __restrict__ e, const float* __restrict__ loss,
              float* __restrict__ scal, int N) {
  __shared__ float r1[1024], r2[1024];
  const int t = threadIdx.x;
  float a = 0.f, b = 0.f;
  for (int i = t; i < N; i += 1024) { a += e[i]; b += loss[i]; }
  r1[t] = a; r2[t] = b; __syncthreads();
  for (int s = 512; s > 0; s >>= 1) {
    if (t < s) { r1[t] += r1[t + s]; r2[t] += r2[t + s]; }
    __syncthreads();
  }
  if (t == 0) { scal[0] = r1[0]; scal[1] = r2[0]; }
}

// ---------------- K9: final scalar -----------------------------------------
__global__ void __launch_bounds__(1024)
final_kernel(const float* __restrict__ w, const float* __restrict__ scal,
             const int* __restrict__ lam_p, float* __restrict__ out, int N) {
  __shared__ float r1[1024], r2[1024];
  const int t = threadIdx.x;
  const float S = scal[0];
  const float lam = (float)lam_p[0];
  const float scale = (float)N / S;
  float a = 0.f, b = 0.f;
  for (int i = t; i < N; i += 1024) {
    float el = w[i] * scale;
    float ex = expf(el / lam);
    a += ex;
    b += ex * el;
  }
  r1[t] = a; r2[t] = b; __syncthreads();
  for (int s = 512; s > 0; s >>= 1) {
    if (t < s) { r1[t] += r1[t + s]; r2[t] += r2[t + s]; }
    __syncthreads();
  }
  if (t == 0) {
    // result = sum(ex*el) / (u*N),  u = GAMMA*mean(ex)  ->  sProd/(GAMMA*sExp)
    out[0] = (lam_p[0] >= 200) ? (scal[1] / (float)N)
                               : (r2[0] / (GAMMA_C * r1[0]));
  }
}

// ---------------------------------------------------------------------------
extern "C" void kernel_launch(void* const* d_in, const int* in_sizes, int n_in,
                              void* d_out, int out_size, void* d_ws, size_t ws_size,
                              hipStream_t stream) {
  const float* logits = (const float*)d_in[0];
  const int*   target = (const int*)d_in[1];
  const float* cw     = (const float*)d_in[2];
  const int*   lam    = (const int*)d_in[3];
  const float* embed  = (const float*)d_in[4];

  const int N = in_sizes[1];            // 8192
  const int C = in_sizes[2];            // 1000
  const int D = in_sizes[4] / N;        // 512

  float* ws   = (float*)d_ws;
  float* loss = ws;                         // N
  float* mu   = loss + N;                   // D
  float* G    = mu + D;                     // D*D
  float* v1   = G + (size_t)D * D;          // D
  float* v2   = v1 + D;                     // D
  float* muv  = v2 + D;                     // 2
  float* proj = muv + 2;                    // 2N
  float* nrm  = proj + (size_t)2 * N;       // 2N
  float* evec = nrm + (size_t)2 * N;        // N
  float* wvec = evec + N;                   // N
  float* scal = wvec + N;                   // 2
  unsigned long long* keys = (unsigned long long*)(scal + 2);  // 2 (8B aligned)

  init_kernel<<<1, 32, 0, stream>>>(keys);
  loss_kernel<<<N, 256, 0, stream>>>(logits, target, cw, loss, N, C);
  colmean_kernel<<<(D + 255) / 256, 256, 0, stream>>>(embed, mu, N, D);

  const int tiles = (D / 16) * (D / 16);   // 1024 tiles, 8 waves per block
  gram_kernel<<<tiles / 8, 256, 0, stream>>>(embed, mu, G, N, D);

  pca_kernel<<<1, 512, 0, stream>>>(G, mu, v1, v2, muv, D);
  project_kernel<<<N, 128, 0, stream>>>(embed, v1, v2, muv, proj, keys, N, D);
  normembed_kernel<<<(N + 255) / 256, 256, 0, stream>>>(proj, keys, nrm, N);

  const size_t lds = (size_t)3 * N * sizeof(float);  // 96 KB (<320 KB WGP LDS)
  topk_kernel<<<N / 256, 256, lds, stream>>>(nrm, loss, evec, wvec, N);

  reduce_kernel<<<1, 1024, 0, stream>>>(evec, loss, scal, N);
  final_kernel<<<1, 1024, 0, stream>>>(wvec, scal, lam, (float*)d_out, N);
}